// P2MLoss_50165218017978
// MI455X (gfx1250) — compile-verified
//
#include <hip/hip_runtime.h>

// ---------------------------------------------------------------------------
// P2M loss for MI455X (gfx1250, wave32).
// Chamfer core: V_WMMA_F32_16X16X4_F32 with the spare K-slot folding |p|^2:
//   A row m = (-2gx, -2gy, -2gz, 1),  B col n = (px, py, pz, |p|^2)
//   => D[m][n] = -2*dot(g,p) + |p|^2 ;  dist = D + |g|^2 (per-row broadcast).
// Two GT tiles per wave share each B chunk (2 back-to-back WMMAs), pred loads
// are software-pipelined so the global_load overlaps the matrix/min work.
// ---------------------------------------------------------------------------

typedef __attribute__((ext_vector_type(2))) float v2f;
typedef __attribute__((ext_vector_type(8))) float v8f;

#define NGT 30000
#define GT_TILES (NGT / 16)              // 1875 exact
#define CH_WAVES ((GT_TILES + 1) / 2)    // 938, two tiles per wave

// Order-preserving uint encoding of float for atomicMin-based float min.
__device__ __forceinline__ unsigned encOrd(float f) {
  unsigned u = __float_as_uint(f);
  return (u & 0x80000000u) ? ~u : (u | 0x80000000u);
}
__device__ __forceinline__ float decOrd(unsigned e) {
  unsigned u = (e & 0x80000000u) ? (e & 0x7FFFFFFFu) : ~e;
  return __uint_as_float(u);
}

// Workspace layout (32-bit units):
//  acc[0..2]   dist1 (GT->pred min) sums per level
//  acc[6..8]   edge squared-length sums per level
//  acc[9..11]  laplace diff squared sums per level
//  acc[12..14] move squared sums per level
//  [16 ..)     uint-encoded per-pred-point min distances (156+618+2466)
__global__ void init_ws_kernel(float* acc, unsigned* d2, int nacc, int nd2) {
  int i = blockIdx.x * blockDim.x + threadIdx.x;
  if (i < nacc) acc[i] = 0.0f;
  if (i < nd2)  d2[i]  = 0xFFFFFFFFu;
}

// One wave per PAIR of 16-row GT tiles; loop over pred points in 16-col chunks.
__global__ void chamfer_wmma_kernel(const float* __restrict__ gt,
                                    const float* __restrict__ pr,
                                    int npr,
                                    unsigned* __restrict__ d2enc,
                                    float* __restrict__ d1sum) {
  const int lane = threadIdx.x & 31;
  const int wave = (blockIdx.x * blockDim.x + threadIdx.x) >> 5;
  if (wave >= CH_WAVES) return;               // wave-uniform exit

  const bool hi  = lane >= 16;
  const int  l15 = lane & 15;
  const int  t0  = wave * 2;
  const bool t1v = (t0 + 1) < GT_TILES;       // wave-uniform
  const int  t1  = t1v ? (t0 + 1) : t0;       // last wave duplicates tile

  // GT rows for both tiles (lane l and l+16 load the same row)
  const int m0 = t0 * 16 + l15, m1 = t1 * 16 + l15;
  const float g0x = gt[3*m0], g0y = gt[3*m0+1], g0z = gt[3*m0+2];
  const float g1x = gt[3*m1], g1y = gt[3*m1+1], g1z = gt[3*m1+2];
  const float gn0l = g0x*g0x + g0y*g0y + g0z*g0z;
  const float gn1l = g1x*g1x + g1y*g1y + g1z*g1z;

  // A operands (16x4 f32): lanes 0-15 -> K0,K1 ; lanes 16-31 -> K2,K3.
  // -2 folded into A (f32 WMMA has no A/B NEG); K3 = 1 pairs with B's |p|^2.
  v2f a0, a1;
  a0.x = hi ? (-2.0f * g0z) : (-2.0f * g0x);
  a0.y = hi ? 1.0f          : (-2.0f * g0y);
  a1.x = hi ? (-2.0f * g1z) : (-2.0f * g1x);
  a1.y = hi ? 1.0f          : (-2.0f * g1y);

  // |g|^2 for the 8 C rows this lane covers (row = v + 8*hi)
  float gn0[8], gn1[8];
#pragma unroll
  for (int v = 0; v < 8; ++v) {
    gn0[v] = __shfl(gn0l, v + (hi ? 8 : 0), 32);
    gn1[v] = __shfl(gn1l, v + (hi ? 8 : 0), 32);
  }

  float r0[8], r1[8];
#pragma unroll
  for (int v = 0; v < 8; ++v) { r0[v] = 3.4e38f; r1[v] = 3.4e38f; }

  const int chunks = (npr + 15) >> 4;

  // Preload chunk 0 pred coords (clamped index -> no divergence)
  float px, py, pz;
  {
    const int ni = (l15 < npr) ? l15 : (npr - 1);
    px = pr[3*ni]; py = pr[3*ni+1]; pz = pr[3*ni+2];
  }

  for (int c = 0; c < chunks; ++c) {
    const int  ncol = c * 16 + l15;
    const bool cv   = ncol < npr;
    const float pn  = cv ? (px*px + py*py + pz*pz) : 1.0e30f;

    // B operand (4x16 f32): lane gives column N; halves give K pairs.
    v2f b;
    b.x = hi ? pz : px;
    b.y = hi ? pn : py;

    // Prefetch next chunk's coords; first use is next iteration, so the
    // load overlaps the WMMAs + min work below.
    {
      const int cn = (c + 1 < chunks) ? (c + 1) : c;
      int n2 = cn * 16 + l15;
      const int ni = (n2 < npr) ? n2 : (npr - 1);
      px = pr[3*ni]; py = pr[3*ni+1]; pz = pr[3*ni+2];
    }

    v8f cz = {0.f, 0.f, 0.f, 0.f, 0.f, 0.f, 0.f, 0.f};
    // D = -2*dot(g,p) + |p|^2 ; two tiles share one B
    v8f d0 = __builtin_amdgcn_wmma_f32_16x16x4_f32(false, a0, false, b,
                                                   (short)0, cz, false, false);
    v8f d1 = __builtin_amdgcn_wmma_f32_16x16x4_f32(false, a1, false, b,
                                                   (short)0, cz, false, false);

    float cmin = 3.4e38f;
#pragma unroll
    for (int v = 0; v < 8; ++v) {
      const float e0 = d0[v] + gn0[v];
      const float e1 = d1[v] + gn1[v];
      r0[v] = fminf(r0[v], e0);
      r1[v] = fminf(r1[v], e1);
      cmin = fminf(cmin, fminf(e0, e1));
    }
    // column min across the two 8-row halves (lane l vs l+16, same column);
    // covers all 32 GT rows of this wave -> one atomic per column per chunk.
    cmin = fminf(cmin, __shfl_xor(cmin, 16, 32));
    if (!hi && cv) atomicMin(&d2enc[ncol], encOrd(cmin));
  }

  // row-wise min across the 16 lanes of each half (rows v / v+8)
#pragma unroll
  for (int v = 0; v < 8; ++v) {
    float r = r0[v];
    r = fminf(r, __shfl_xor(r, 8, 32));
    r = fminf(r, __shfl_xor(r, 4, 32));
    r = fminf(r, __shfl_xor(r, 2, 32));
    r = fminf(r, __shfl_xor(r, 1, 32));
    r0[v] = r;
    float s = r1[v];
    s = fminf(s, __shfl_xor(s, 8, 32));
    s = fminf(s, __shfl_xor(s, 4, 32));
    s = fminf(s, __shfl_xor(s, 2, 32));
    s = fminf(s, __shfl_xor(s, 1, 32));
    r1[v] = s;
  }
  if (lane == 0 || lane == 16) {
    float s = 0.f;
#pragma unroll
    for (int v = 0; v < 8; ++v) s += r0[v];
    if (t1v) {                       // skip duplicated tile's row sums
#pragma unroll
      for (int v = 0; v < 8; ++v) s += r1[v];
    }
    atomicAdd(d1sum, s);
  }
}

__global__ void edge_kernel(const float* __restrict__ pred,
                            const int* __restrict__ edges, int ne,
                            float* __restrict__ esum) {
  int i = blockIdx.x * blockDim.x + threadIdx.x;
  float s = 0.f;
  if (i < ne) {
    int a = edges[2 * i], b = edges[2 * i + 1];
    float dx = pred[3 * a]     - pred[3 * b];
    float dy = pred[3 * a + 1] - pred[3 * b + 1];
    float dz = pred[3 * a + 2] - pred[3 * b + 2];
    s = dx * dx + dy * dy + dz * dz;
  }
  for (int off = 16; off; off >>= 1) s += __shfl_xor(s, off, 32);
  if ((threadIdx.x & 31) == 0) atomicAdd(esum, s);
}

__global__ void lap_kernel(const float* __restrict__ pbd,
                           const float* __restrict__ pred,
                           const int* __restrict__ lap, int n,
                           float* __restrict__ lapsum,
                           float* __restrict__ movesum) {
  int i = blockIdx.x * blockDim.x + threadIdx.x;
  float sq = 0.f, mq = 0.f;
  if (i < n) {
    float bx = pbd[3 * i], by = pbd[3 * i + 1], bz = pbd[3 * i + 2];
    float ax = pred[3 * i], ay = pred[3 * i + 1], az = pred[3 * i + 2];
    float sbx = 0, sby = 0, sbz = 0, sax = 0, say = 0, saz = 0;
#pragma unroll
    for (int k = 0; k < 8; ++k) {
      int nb = lap[10 * i + k];
      if (nb >= 0) {
        sbx += pbd[3 * nb]; sby += pbd[3 * nb + 1]; sbz += pbd[3 * nb + 2];
        sax += pred[3 * nb]; say += pred[3 * nb + 1]; saz += pred[3 * nb + 2];
      }
    }
    float deg = (float)lap[10 * i + 9];
    float dx = (bx - sbx / deg) - (ax - sax / deg);
    float dy = (by - sby / deg) - (ay - say / deg);
    float dz = (bz - sbz / deg) - (az - saz / deg);
    sq = dx * dx + dy * dy + dz * dz;
    float mx = bx - ax, my = by - ay, mz = bz - az;
    mq = mx * mx + my * my + mz * mz;
  }
  for (int off = 16; off; off >>= 1) {
    sq += __shfl_xor(sq, off, 32);
    mq += __shfl_xor(mq, off, 32);
  }
  if ((threadIdx.x & 31) == 0) {
    atomicAdd(lapsum, sq);
    atomicAdd(movesum, mq);
  }
}

__global__ void finalize_kernel(const float* __restrict__ acc,
                                const unsigned* __restrict__ d2base,
                                float* __restrict__ out) {
  __shared__ float sh[256];
  __shared__ float d2s[3];
  const int NVs[3] = {156, 618, 2466};
  int off = 0;
  for (int i = 0; i < 3; ++i) {
    float p = 0.f;
    for (int j = threadIdx.x; j < NVs[i]; j += 256) p += decOrd(d2base[off + j]);
    sh[threadIdx.x] = p;
    __syncthreads();
    for (int s = 128; s; s >>= 1) {
      if ((int)threadIdx.x < s) sh[threadIdx.x] += sh[threadIdx.x + s];
      __syncthreads();
    }
    if (threadIdx.x == 0) d2s[i] = sh[0];
    __syncthreads();
    off += NVs[i];
  }
  if (threadIdx.x == 0) {
    const float NEs[3]  = {462.f, 1848.f, 7392.f};
    const float LAPC[3] = {0.2f, 1.f, 1.f};
    float chamfer = 0.f, edge = 0.f, lap = 0.f;
    for (int i = 0; i < 3; ++i) {
      chamfer += acc[i] / 30000.f + d2s[i] / (float)NVs[i];
      edge += acc[6 + i] / NEs[i] * 300.f;
      // mean(d*d)*3 == sum/(3n)*3 == sum/n ; same for move term
      float l = acc[9 + i] / (float)NVs[i];
      if (i > 0) l += acc[12 + i] / (float)NVs[i];
      lap += LAPC[i] * l;
    }
    out[0] = 100.f * chamfer + 0.1f * edge + 0.3f * lap;
    out[1] = chamfer;
    out[2] = edge;
    out[3] = lap;
  }
}

extern "C" void kernel_launch(void* const* d_in, const int* in_sizes, int n_in,
                              void* d_out, int out_size, void* d_ws, size_t ws_size,
                              hipStream_t stream) {
  // dict order: points, then per level i: pred_coord_i, pbd_i, edges_i, lap_idx_i
  const float* points = (const float*)d_in[0];
  const int NVs[3] = {156, 618, 2466};
  const int NEs[3] = {462, 1848, 7392};

  float* acc = (float*)d_ws;
  unsigned* d2 = (unsigned*)d_ws + 16;
  const int totd2 = 156 + 618 + 2466;

  init_ws_kernel<<<(16 + totd2 + 255) / 256, 256, 0, stream>>>(acc, d2, 16, totd2);

  int off = 0;
  for (int i = 0; i < 3; ++i) {
    const float* pred  = (const float*)d_in[1 + 4 * i];
    const float* pbd   = (const float*)d_in[2 + 4 * i];
    const int*   edges = (const int*)d_in[3 + 4 * i];
    const int*   lap   = (const int*)d_in[4 + 4 * i];
    const int n = NVs[i], e = NEs[i];

    const int blocks = (CH_WAVES + 7) / 8;  // 8 waves / 256-thread block
    chamfer_wmma_kernel<<<blocks, 256, 0, stream>>>(points, pbd, n, d2 + off, acc + i);
    edge_kernel<<<(e + 255) / 256, 256, 0, stream>>>(pred, edges, e, acc + 6 + i);
    lap_kernel<<<(n + 255) / 256, 256, 0, stream>>>(pbd, pred, lap, n,
                                                    acc + 9 + i, acc + 12 + i);
    off += n;
  }
  finalize_kernel<<<1, 256, 0, stream>>>(acc, d2, (float*)d_out);
}